// spatial_encoding_block_29686813950259
// MI455X (gfx1250) — compile-verified
//
#include <hip/hip_runtime.h>
#include <hip/hip_bf16.h>
#include <stdint.h>

// ---------------------------------------------------------------------------
// spatial_encoding_block: nearest-fill + depthwise 2x2 conv + leaky + maxpool
// HBM-bound stream kernel (AI ~1.25 FLOP/B, floor ~12us @ 23.3 TB/s).
// CDNA5 async global->LDS B128 staging (ASYNCcnt); all 9 tile loads share one
// VADDR/VDST and use the 24-bit instruction offset (applies to both sides).
// 32-bit indexing throughout (x = 247 MB < 2 GB).
// ---------------------------------------------------------------------------

#define THREADS          256
#define PAIRS_PER_BLOCK  1024
#define PPT              (PAIRS_PER_BLOCK / THREADS)   // 4 pairs per thread
#define XDW_PER_BLOCK    (PAIRS_PER_BLOCK * 9)         // 9216 dwords = 36 KB
#define C_CH             105

typedef int v4i_t __attribute__((__vector_size__(16)));
typedef __attribute__((address_space(1))) v4i_t gv4i_t;
typedef __attribute__((address_space(3))) v4i_t lv4i_t;

template <int OFF>
__device__ __forceinline__ void async_copy_b128(const float* gsrc, float* lds_dst) {
#if __has_builtin(__builtin_amdgcn_global_load_async_to_lds_b128)
    __builtin_amdgcn_global_load_async_to_lds_b128(
        (gv4i_t*)(void*)(float*)gsrc, (lv4i_t*)(void*)lds_dst, OFF, /*cpol*/0);
#else
    unsigned loff = (unsigned)(unsigned long long)
        (__attribute__((address_space(3))) void*)lds_dst;
    asm volatile("global_load_async_to_lds_b128 %0, %1, off offset:%2"
                 :: "v"(loff), "v"(gsrc), "i"(OFF) : "memory");
#endif
}

__device__ __forceinline__ void wait_async_all() {
#if __has_builtin(__builtin_amdgcn_s_wait_asynccnt)
    __builtin_amdgcn_s_wait_asynccnt(0);
#else
    asm volatile("s_wait_asynccnt 0" ::: "memory");
#endif
}

__global__ __launch_bounds__(THREADS)
void spatial_encoding_block_kernel(const float* __restrict__ x,
                                   const float* __restrict__ w,
                                   const float* __restrict__ b,
                                   float* __restrict__ out,
                                   unsigned n_pairs) {
    __shared__ float  s_x[XDW_PER_BLOCK];   // 36864 B x-tile
    __shared__ float4 s_w[C_CH];            // depthwise 2x2 weights
    __shared__ float  s_b[C_CH];            // bias
    __shared__ int    s_nearb[9];           // fill map as BYTE offsets (k*4 if kept)

    const int      tid     = threadIdx.x;
    const unsigned pbase   = blockIdx.x * PAIRS_PER_BLOCK;
    const unsigned base_dw = pbase * 9u;                 // dword offset of tile

    // ---- Phase 1: async-stage the x tile; one VADDR/VDST, 9 imm offsets ----
    const float* gsrc = x + (base_dw + (unsigned)tid * 4u);
    float*       ldst = &s_x[tid * 4];
    if (pbase + PAIRS_PER_BLOCK <= n_pairs) {            // full tile (always, here)
        async_copy_b128<0 * 4096>(gsrc, ldst);
        async_copy_b128<1 * 4096>(gsrc, ldst);
        async_copy_b128<2 * 4096>(gsrc, ldst);
        async_copy_b128<3 * 4096>(gsrc, ldst);
        async_copy_b128<4 * 4096>(gsrc, ldst);
        async_copy_b128<5 * 4096>(gsrc, ldst);
        async_copy_b128<6 * 4096>(gsrc, ldst);
        async_copy_b128<7 * 4096>(gsrc, ldst);
        async_copy_b128<8 * 4096>(gsrc, ldst);
    } else {                                             // partial tail block
        const unsigned maxdw = n_pairs * 9u - 4u;        // last aligned float4
#pragma unroll
        for (int k = 0; k < 9; ++k) {
            unsigned gdw = base_dw + (unsigned)tid * 4u + (unsigned)k * 1024u;
            if (gdw > maxdw) gdw = maxdw;
            async_copy_b128<0>(x + gdw, &s_x[tid * 4 + k * 1024]);
        }
    }

    // ---- Phase 2 (overlapped): stage weights + bias, compute fill map ----
    if (tid < C_CH) {
        s_w[tid] = ((const float4*)w)[tid];              // one B128 load/store
        s_b[tid] = b[tid];
    }
    if (tid == 0) {
        // Mask & nearest-source map come from sample 0, channel 0 only.
        float p0[9];
#pragma unroll
        for (int k = 0; k < 9; ++k) p0[k] = x[k];
#pragma unroll
        for (int k = 0; k < 9; ++k) {
            int nk = k;
            if (p0[k] == 0.0f) {
                int best = 0x7fffffff, bi = 0;
#pragma unroll
                for (int s = 0; s < 9; ++s) {
                    int di = (k / 3) - (s / 3);
                    int dj = (k % 3) - (s % 3);
                    int d  = (p0[s] == 0.0f) ? 0x7fffffff : (di * di + dj * dj);
                    if (d < best) { best = d; bi = s; }   // strict '<' => argmin tie
                }                                         //   breaks to lowest index
                nk = bi;
            }
            s_nearb[k] = nk * 4;                          // byte offset
        }
    }

    wait_async_all();      // our wave's async tile writes landed in LDS
    __syncthreads();       // everyone else's too

    int nearb[9];
#pragma unroll
    for (int k = 0; k < 9; ++k) nearb[k] = s_nearb[k];

    const unsigned c0 = pbase % C_CH;                     // block-uniform (scalar)

    // ---- Phase 3: per-pair compute; stride-9 LDS reads are conflict-free ----
#pragma unroll
    for (int j = 0; j < PPT; ++j) {
        const int      pl = tid + j * THREADS;            // pair within block
        const unsigned p  = pbase + (unsigned)pl;
        if (p >= n_pairs) continue;

        const char* x9b = (const char*)(const void*)s_x + pl * 36;
        float a[9];
#pragma unroll
        for (int k = 0; k < 9; ++k)                       // gather = original vals
            a[k] = *(const float*)(x9b + nearb[k]);

        // c = (c0 + pl) % 105, exact for t < ~4096 via magic 39946/2^22
        const unsigned t = c0 + (unsigned)pl;
        const unsigned c = t - 105u * ((t * 39946u) >> 22);

        const float4 wc = s_w[c];
        const float  bc = s_b[c];

        // depthwise 2x2 valid conv over the filled 3x3, + bias
        float y00 = fmaf(a[0], wc.x, fmaf(a[1], wc.y, fmaf(a[3], wc.z, fmaf(a[4], wc.w, bc))));
        float y01 = fmaf(a[1], wc.x, fmaf(a[2], wc.y, fmaf(a[4], wc.z, fmaf(a[5], wc.w, bc))));
        float y10 = fmaf(a[3], wc.x, fmaf(a[4], wc.y, fmaf(a[6], wc.z, fmaf(a[7], wc.w, bc))));
        float y11 = fmaf(a[4], wc.x, fmaf(a[5], wc.y, fmaf(a[7], wc.z, fmaf(a[8], wc.w, bc))));

        // LeakyReLU(0.2)
        y00 = y00 > 0.0f ? y00 : 0.2f * y00;
        y01 = y01 > 0.0f ? y01 : 0.2f * y01;
        y10 = y10 > 0.0f ? y10 : 0.2f * y10;
        y11 = y11 > 0.0f ? y11 : 0.2f * y11;

        // MaxPool2d(2) over the 2x2, then add filled center pixel
        const float m = fmaxf(fmaxf(y00, y01), fmaxf(y10, y11));
        out[p] = m + a[4];
    }
}

extern "C" void kernel_launch(void* const* d_in, const int* in_sizes, int n_in,
                              void* d_out, int out_size, void* d_ws, size_t ws_size,
                              hipStream_t stream) {
    (void)n_in; (void)d_ws; (void)ws_size; (void)in_sizes;
    const float* x = (const float*)d_in[0];   // [B,105,3,3] f32
    const float* w = (const float*)d_in[1];   // [105,1,2,2] f32
    const float* b = (const float*)d_in[2];   // [105]       f32
    float* out = (float*)d_out;               // [B*105]     f32

    const unsigned n_pairs = (unsigned)out_size;   // B*C = 6,881,280 = 6720*1024
    const int nblocks = (int)((n_pairs + PAIRS_PER_BLOCK - 1) / PAIRS_PER_BLOCK);

    spatial_encoding_block_kernel<<<nblocks, THREADS, 0, stream>>>(x, w, b, out, n_pairs);
}